// Code_projection_17824114278457
// MI455X (gfx1250) — compile-verified
//
#include <hip/hip_runtime.h>
#include <math.h>

#define Bb 2
#define Dd 512
#define Kk 32
#define Nn 4096   // H*W = 64*64

typedef __attribute__((ext_vector_type(2))) float v2f;
typedef __attribute__((ext_vector_type(8))) float v8f;

__device__ __forceinline__ v8f wmma4(v2f a, v2f b, v8f c) {
  // D = A(16x4 f32) * B(4x16 f32) + C(16x16 f32)
  return __builtin_amdgcn_wmma_f32_16x16x4_f32(
      /*neg_a=*/false, a, /*neg_b=*/false, b,
      /*c_mod=*/(short)0, c, /*reuse_a=*/false, /*reuse_b=*/false);
}

// ds_swizzle group-of-32: offset[4:0]=and=0x1f, offset[9:5]=or=0, offset[14:10]=xor
#define SWZ_XOR(v, m) \
  __int_as_float(__builtin_amdgcn_ds_swizzle(__float_as_int(v), (0x1f | ((m) << 10))))

// Softmax across the 32 codewords for one pixel row. v0 holds this lane's
// logit for k=laneM, v1 for k=16+laneM; the other 15 k live in the other
// lanes of this 16-lane half (xor masks 1..8 never cross halves).
__device__ __forceinline__ void softmax_row(float v0, float v1,
                                            float& q0, float& q1) {
  float m = fmaxf(v0, v1);
  m = fmaxf(m, SWZ_XOR(m, 1));
  m = fmaxf(m, SWZ_XOR(m, 2));
  m = fmaxf(m, SWZ_XOR(m, 4));
  m = fmaxf(m, SWZ_XOR(m, 8));
  float e0 = __expf(v0 - m);
  float e1 = __expf(v1 - m);
  float s = e0 + e1;
  s += SWZ_XOR(s, 1);
  s += SWZ_XOR(s, 2);
  s += SWZ_XOR(s, 4);
  s += SWZ_XOR(s, 8);
  float inv = 1.0f / s;
  q0 = e0 * inv;
  q1 = e1 * inv;
}

// ---------------- prep: Wa = s^2, Wb = -2 s^2 c, cterm[k] = sum_d s^2 c^2 ----
__global__ void kprep(const float* __restrict__ codewords,
                      const float* __restrict__ scale,
                      float* __restrict__ Wa, float* __restrict__ Wb) {
  int i = blockIdx.x * blockDim.x + threadIdx.x;
  if (i >= Dd * Kk) return;
  int d = i / Kk, k = i % Kk;
  float s  = scale[i];
  float s2 = s * s;
  Wa[i] = s2;
  Wb[i] = -2.0f * s2 * codewords[k * Dd + d];
}

__global__ void kcterm(const float* __restrict__ codewords,
                       const float* __restrict__ scale,
                       float* __restrict__ cterm) {
  __shared__ float red[256];
  int k = blockIdx.x;
  int tid = threadIdx.x;
  float acc = 0.0f;
  for (int d = tid; d < Dd; d += 256) {
    float s = scale[d * Kk + k];
    float c = codewords[k * Dd + d];
    acc += s * s * c * c;
  }
  red[tid] = acc; __syncthreads();
  for (int off = 128; off > 0; off >>= 1) {
    if (tid < off) red[tid] += red[tid + off];
    __syncthreads();
  }
  if (tid == 0) cterm[k] = red[0];
}

// ---------------- logits GEMM + softmax over K ------------------------------
// One wave owns 32 pixels (two 16-row M tiles) x all 32 codewords (two
// 16-col K tiles): 4 accumulators, B operands shared across both M tiles.
__global__ void klogits(const float* __restrict__ X,
                        const float* __restrict__ Wa,
                        const float* __restrict__ Wb,
                        const float* __restrict__ cterm,
                        float* __restrict__ Qout) {
  const int tilesPerBatch = Nn / 32;                 // 128
  int wave = threadIdx.x >> 5;
  int tile = blockIdx.x * (blockDim.x >> 5) + wave;
  if (tile >= Bb * tilesPerBatch) return;            // wave-uniform
  int b  = tile / tilesPerBatch;
  int n0 = (tile % tilesPerBatch) * 32;
  int lane  = threadIdx.x & 31;
  int laneM = lane & 15;
  int laneH = lane >> 4;

  const float* Xb = X + (size_t)b * Dd * Nn;
  v8f accA0 = {};   // pixels n0..n0+15,  k 0..15
  v8f accA1 = {};   // pixels n0..n0+15,  k 16..31
  v8f accB0 = {};   // pixels n0+16..+31, k 0..15
  v8f accB1 = {};   // pixels n0+16..+31, k 16..31

  #pragma unroll 2
  for (int dd = 0; dd < Dd; dd += 4) {
    int di = dd + 2 * laneH;
    v2f aA, aA2, aB, aB2, ba0, ba1, bb0, bb1;
    aA.x = Xb[(size_t)di * Nn + n0 + laneM];
    aA.y = Xb[(size_t)(di + 1) * Nn + n0 + laneM];
    aB.x = Xb[(size_t)di * Nn + n0 + 16 + laneM];
    aB.y = Xb[(size_t)(di + 1) * Nn + n0 + 16 + laneM];
    aA2.x = aA.x * aA.x; aA2.y = aA.y * aA.y;
    aB2.x = aB.x * aB.x; aB2.y = aB.y * aB.y;
    ba0.x = Wa[di * Kk + laneM];          ba0.y = Wa[(di + 1) * Kk + laneM];
    ba1.x = Wa[di * Kk + 16 + laneM];     ba1.y = Wa[(di + 1) * Kk + 16 + laneM];
    bb0.x = Wb[di * Kk + laneM];          bb0.y = Wb[(di + 1) * Kk + laneM];
    bb1.x = Wb[di * Kk + 16 + laneM];     bb1.y = Wb[(di + 1) * Kk + 16 + laneM];
    accA0 = wmma4(aA2, ba0, accA0);   // + s^2 x^2
    accA0 = wmma4(aA,  bb0, accA0);   // - 2 s^2 c x
    accA1 = wmma4(aA2, ba1, accA1);
    accA1 = wmma4(aA,  bb1, accA1);
    accB0 = wmma4(aB2, ba0, accB0);
    accB0 = wmma4(aB,  bb0, accB0);
    accB1 = wmma4(aB2, ba1, accB1);
    accB1 = wmma4(aB,  bb1, accB1);
  }

  float ct0 = cterm[laneM];
  float ct1 = cterm[16 + laneM];
  // D tile layout: vgpr r, lanes 0-15 -> row r, lanes 16-31 -> row r+8; col = laneM
  #pragma unroll
  for (int r = 0; r < 8; ++r) {
    float q0, q1;
    softmax_row(-0.5f * (accA0[r] + ct0), -0.5f * (accA1[r] + ct1), q0, q1);
    int row = n0 + r + 8 * laneH;
    float* qp = Qout + (size_t)(b * Nn + row) * Kk;
    qp[laneM]      = q0;
    qp[16 + laneM] = q1;
  }
  #pragma unroll
  for (int r = 0; r < 8; ++r) {
    float q0, q1;
    softmax_row(-0.5f * (accB0[r] + ct0), -0.5f * (accB1[r] + ct1), q0, q1);
    int row = n0 + 16 + r + 8 * laneH;
    float* qp = Qout + (size_t)(b * Nn + row) * Kk;
    qp[laneM]      = q0;
    qp[16 + laneM] = q1;
  }
}

// ---------------- qsum[b,k] = sum_n Q (deterministic block reduction) -------
__global__ void kqsum(const float* __restrict__ Q, float* __restrict__ qsum) {
  __shared__ float red[256];
  int bk = blockIdx.x;                 // 0..B*K-1
  int b = bk / Kk, k = bk % Kk;
  int tid = threadIdx.x;
  float acc = 0.0f;
  for (int n = tid; n < Nn; n += 256) acc += Q[(size_t)(b * Nn + n) * Kk + k];
  red[tid] = acc; __syncthreads();
  for (int off = 128; off > 0; off >>= 1) {
    if (tid < off) red[tid] += red[tid + off];
    __syncthreads();
  }
  if (tid == 0) qsum[bk] = red[0];
}

// ---------------- aggregation GEMM: QX[b,d,k] = sum_n X[b,d,n]*Q[b,n,k] -----
// One wave owns two 16-row D tiles (B operand = Q tile shared), then
// Zraw = scale * (QX/qsum - codewords).
__global__ void kagg(const float* __restrict__ X,
                     const float* __restrict__ Q,
                     const float* __restrict__ qsum,
                     const float* __restrict__ codewords,
                     const float* __restrict__ scale,
                     float* __restrict__ Zraw) {
  const int tilesPerBatch = (Dd / 32) * (Kk / 16);   // 32
  int wave = threadIdx.x >> 5;
  int tile = blockIdx.x * (blockDim.x >> 5) + wave;
  if (tile >= Bb * tilesPerBatch) return;            // wave-uniform
  int b   = tile / tilesPerBatch;
  int rem = tile % tilesPerBatch;
  int d0 = (rem >> 1) * 32;
  int k0 = (rem & 1) * 16;
  int lane  = threadIdx.x & 31;
  int laneM = lane & 15;
  int laneH = lane >> 4;

  const float* Xrow0 = X + (size_t)b * Dd * Nn + (size_t)(d0 + laneM) * Nn;
  const float* Xrow1 = Xrow0 + (size_t)16 * Nn;
  const float* Qb    = Q + (size_t)b * Nn * Kk;
  v8f acc0 = {};    // d0..d0+15
  v8f acc1 = {};    // d0+16..d0+31

  #pragma unroll 4
  for (int nn = 0; nn < Nn; nn += 4) {
    int ni = nn + 2 * laneH;
    v2f a0, a1, bv;
    a0.x = Xrow0[ni];  a0.y = Xrow0[ni + 1];   // contiguous -> b64 load
    a1.x = Xrow1[ni];  a1.y = Xrow1[ni + 1];
    bv.x = Qb[(size_t)ni * Kk + k0 + laneM];
    bv.y = Qb[(size_t)(ni + 1) * Kk + k0 + laneM];
    acc0 = wmma4(a0, bv, acc0);
    acc1 = wmma4(a1, bv, acc1);
  }

  int k = k0 + laneM;
  float invqs = 1.0f / qsum[b * Kk + k];
  #pragma unroll
  for (int r = 0; r < 8; ++r) {
    int d = d0 + r + 8 * laneH;
    float zr = scale[d * Kk + k] * (acc0[r] * invqs - codewords[k * Dd + d]);
    Zraw[(size_t)(b * Dd + d) * Kk + k] = zr;
  }
  #pragma unroll
  for (int r = 0; r < 8; ++r) {
    int d = d0 + 16 + r + 8 * laneH;
    float zr = scale[d * Kk + k] * (acc1[r] * invqs - codewords[k * Dd + d]);
    Zraw[(size_t)(b * Dd + d) * Kk + k] = zr;
  }
}

// ---------------- normsq[b,k] = sum_d Zraw^2 --------------------------------
__global__ void knormsq(const float* __restrict__ Zraw, float* __restrict__ normsq) {
  __shared__ float red[256];
  int bk = blockIdx.x;
  int b = bk / Kk, k = bk % Kk;
  int tid = threadIdx.x;
  float acc = 0.0f;
  for (int d = tid; d < Dd; d += 256) {
    float z = Zraw[(size_t)(b * Dd + d) * Kk + k];
    acc += z * z;
  }
  red[tid] = acc; __syncthreads();
  for (int off = 128; off > 0; off >>= 1) {
    if (tid < off) red[tid] += red[tid + off];
    __syncthreads();
  }
  if (tid == 0) normsq[bk] = red[0];
}

// ---------------- Z = Zraw * rsqrt(normsq) ----------------------------------
__global__ void kfinal(const float* __restrict__ Zraw,
                       const float* __restrict__ normsq,
                       float* __restrict__ Z) {
  int i = blockIdx.x * blockDim.x + threadIdx.x;
  if (i >= Bb * Dd * Kk) return;
  int b = i / (Dd * Kk);
  int k = i % Kk;
  Z[i] = Zraw[i] * rsqrtf(normsq[b * Kk + k]);
}

extern "C" void kernel_launch(void* const* d_in, const int* in_sizes, int n_in,
                              void* d_out, int out_size, void* d_ws, size_t ws_size,
                              hipStream_t stream) {
  const float* X         = (const float*)d_in[0];   // [B, D, H, W]
  const float* codewords = (const float*)d_in[1];   // [K, D]
  const float* scale     = (const float*)d_in[2];   // [D, K]

  float* out = (float*)d_out;
  float* Z = out;                       // [B, D, K]  (first tuple element)
  float* Q = out + Bb * Dd * Kk;        // [B, N, K]  (second tuple element)

  float* ws     = (float*)d_ws;
  float* Wa     = ws;                   // D*K
  float* Wb     = Wa + Dd * Kk;         // D*K
  float* cterm  = Wb + Dd * Kk;         // K
  float* qsum   = cterm + Kk;           // B*K
  float* normsq = qsum + Bb * Kk;       // B*K
  float* Zraw   = normsq + Bb * Kk;     // B*D*K

  kprep  <<<(Dd * Kk + 255) / 256, 256, 0, stream>>>(codewords, scale, Wa, Wb);
  kcterm <<<Kk, 256, 0, stream>>>(codewords, scale, cterm);
  klogits<<<(Bb * (Nn / 32)) / 4, 128, 0, stream>>>(X, Wa, Wb, cterm, Q);
  kqsum  <<<Bb * Kk, 256, 0, stream>>>(Q, qsum);
  kagg   <<<(Bb * (Dd / 32) * (Kk / 16)) / 4, 128, 0, stream>>>(X, Q, qsum, codewords, scale, Zraw);
  knormsq<<<Bb * Kk, 256, 0, stream>>>(Zraw, normsq);
  kfinal <<<(Bb * Dd * Kk + 255) / 256, 256, 0, stream>>>(Zraw, normsq, Z);
}